// CamAwareSCLHead_80101140070638
// MI455X (gfx1250) — compile-verified
//
#include <hip/hip_runtime.h>

// CDNA5 / gfx1250, wave32. Fused two-pass cam-aware SupCon loss.
//   pack    : features[N,2,D] -> f[2N,D] in workspace
//   count   : per-row positive counts (labels only)
//   gemm<0> : f f^T tiles via V_WMMA_F32_16X16X4_F32, online row max + 2 neg-sums
//   gemm<1> : recompute tiles, per-positive CE terms, per-row loss
//   reduce  : deterministic tree sum -> scalar
// v2: each wave computes a 16x64 tile (4 independent WMMA accumulator chains,
//     A-fragment reuse x4), batched max-rescale in pass A, branchy sparse
//     positives in pass B, global_prefetch of next B chunk.

constexpr int kN   = 4096;   // samples
constexpr int kD   = 256;    // feature dim
constexpr int kN2  = 8192;   // 2N rows/cols
constexpr float kInvTemp = 10.0f;   // 1/TEMP
constexpr float kNegInf  = -1e30f;
constexpr int kLDP = 260;    // padded LDS row stride (floats): 16B-aligned, bank-safe
constexpr int kRowsWG = 128; // rows per workgroup (8 waves x 16 rows)
constexpr int kCT = 64;      // column tile per iteration (4 sub-tiles of 16)

typedef float v2f __attribute__((ext_vector_type(2)));
typedef float v8f __attribute__((ext_vector_type(8)));

// ---------------- pack: [N,2,D] -> [2N,D] ----------------
__global__ __launch_bounds__(256) void pack_kernel(const float* __restrict__ feat,
                                                   float* __restrict__ f) {
  int idx = blockIdx.x * blockDim.x + threadIdx.x;   // over float4s, exactly kN2*kD/4
  int r = idx >> 6;                                  // kD/4 = 64 float4 per row
  int c = idx & 63;
  int i = r & (kN - 1);
  int s = r >> 12;                                   // r / kN
  float4 v = ((const float4*)(feat + ((size_t)i * 2 + s) * kD))[c];
  ((float4*)(f + (size_t)r * kD))[c] = v;
}

// ---------------- positive counts (labels only) ----------------
__global__ __launch_bounds__(256) void count_kernel(const int* __restrict__ label,
                                                    const int* __restrict__ camid,
                                                    float* __restrict__ p1,
                                                    float* __restrict__ p2) {
  __shared__ int lab[kN];
  __shared__ int cam[kN];
  for (int t = threadIdx.x; t < kN; t += blockDim.x) { lab[t] = label[t]; cam[t] = camid[t]; }
  __syncthreads();
  int i = blockIdx.x * blockDim.x + threadIdx.x;
  int li = lab[i], ci = cam[i];
  int c1 = 0, c2 = 0;
  for (int j = 0; j < kN; ++j) {
    bool le = (lab[j] == li);
    c1 += le ? 1 : 0;
    c2 += (le && cam[j] == ci) ? 1 : 0;
  }
  float n1 = 2.0f * (float)c1 - 1.0f;   // tiled 2x2, diagonal excluded
  float n2 = 2.0f * (float)c2 - 1.0f;
  p1[i] = n1; p1[i + kN] = n1;
  p2[i] = n2; p2[i + kN] = n2;
}

// ---------------- fused GEMM + row reductions ----------------
template <int PASS>
__global__ __launch_bounds__(256) void scl_gemm_kernel(
    const float* __restrict__ f, const int* __restrict__ label, const int* __restrict__ camid,
    float* __restrict__ mArr, float* __restrict__ s1Arr, float* __restrict__ s2Arr,
    const float* __restrict__ p1Arr, const float* __restrict__ p2Arr,
    float* __restrict__ rowloss) {
  __shared__ float aLds[kRowsWG * kLDP];  // 128 x 256 padded (~130 KB)
  __shared__ float bLds[kCT * kLDP];      // 64 cols x 256 (~65 KB)
  __shared__ int clabS[kCT];
  __shared__ int ccamS[kCT];

  const int tid  = threadIdx.x;
  const int wave = tid >> 5;
  const int lane = tid & 31;
  const int half = lane >> 4;      // wave32 half (0/1)
  const int l15  = lane & 15;
  const int rowBase = blockIdx.x * kRowsWG;
  const int waveRow = wave * 16;

  // Stage A tile (all 8 waves' rows) into LDS.
  for (int t = tid * 4; t < kRowsWG * kD; t += 256 * 4) {
    int r = t >> 8, k = t & (kD - 1);
    float4 v = *(const float4*)(f + (size_t)(rowBase + r) * kD + k);
    *(float4*)(&aLds[r * kLDP + k]) = v;
  }

  // Per-lane row identities for the 8 C-slots (C layout: VGPR k -> M=k+8*half).
  int rg[8], rlab[8], rcam[8];
#pragma unroll
  for (int k = 0; k < 8; ++k) {
    rg[k]   = rowBase + waveRow + k + 8 * half;
    rlab[k] = label[rg[k] & (kN - 1)];
    rcam[k] = camid[rg[k] & (kN - 1)];
  }

  float mx[8], s1[8], s2[8];                       // PASS 0 state
  float acc1[8], acc2[8], mrow[8], n1r[8], n2r[8]; // PASS 1 state
#pragma unroll
  for (int k = 0; k < 8; ++k) {
    mx[k] = kNegInf; s1[k] = 0.f; s2[k] = 0.f;
    acc1[k] = 0.f; acc2[k] = 0.f; mrow[k] = 0.f; n1r[k] = 0.f; n2r[k] = 0.f;
  }
  if (PASS == 1) {
#pragma unroll
    for (int k = 0; k < 8; ++k) {
      mrow[k] = mArr[rg[k]];
      n1r[k]  = s1Arr[rg[k]];
      n2r[k]  = s2Arr[rg[k]];
    }
  }

  for (int ct = 0; ct < kN2 / kCT; ++ct) {
    __syncthreads();  // A ready (first iter) / previous B consumed
    const int cbase = ct * kCT;
    for (int t = tid * 4; t < kCT * kD; t += 256 * 4) {
      int c = t >> 8, k = t & (kD - 1);
      float4 v = *(const float4*)(f + (size_t)(cbase + c) * kD + k);
      *(float4*)(&bLds[c * kLDP + k]) = v;
    }
    if (tid < kCT) {
      clabS[tid] = label[(cbase + tid) & (kN - 1)];
      ccamS[tid] = camid[(cbase + tid) & (kN - 1)];
    }
    // Prefetch next column chunk while we compute this one.
    if (cbase + kCT < kN2)
      __builtin_prefetch(f + (size_t)(cbase + kCT) * kD + tid * 64, 0, 3);
    __syncthreads();

    // 16x64 per wave: 4 independent accumulator chains, A fragment reused x4.
    v8f c0 = {}, c1 = {}, c2 = {}, c3 = {};
#pragma unroll 4
    for (int k0 = 0; k0 < kD; k0 += 4) {
      const int ko = k0 + 2 * half;
      v2f a  = *(const v2f*)(&aLds[(waveRow + l15) * kLDP + ko]);
      v2f b0 = *(const v2f*)(&bLds[(l15)      * kLDP + ko]);
      v2f b1 = *(const v2f*)(&bLds[(16 + l15) * kLDP + ko]);
      v2f b2 = *(const v2f*)(&bLds[(32 + l15) * kLDP + ko]);
      v2f b3 = *(const v2f*)(&bLds[(48 + l15) * kLDP + ko]);
      c0 = __builtin_amdgcn_wmma_f32_16x16x4_f32(false, a, false, b0, (short)0, c0, false, false);
      c1 = __builtin_amdgcn_wmma_f32_16x16x4_f32(false, a, false, b1, (short)0, c1, false, false);
      c2 = __builtin_amdgcn_wmma_f32_16x16x4_f32(false, a, false, b2, (short)0, c2, false, false);
      c3 = __builtin_amdgcn_wmma_f32_16x16x4_f32(false, a, false, b3, (short)0, c3, false, false);
    }

    int cgv[4], cl[4], cc[4];
#pragma unroll
    for (int s = 0; s < 4; ++s) {
      cgv[s] = cbase + s * 16 + l15;
      cl[s]  = clabS[s * 16 + l15];
      cc[s]  = ccamS[s * 16 + l15];
    }

#pragma unroll
    for (int k = 0; k < 8; ++k) {
      float v[4] = { c0[k] * kInvTemp, c1[k] * kInvTemp,
                     c2[k] * kInvTemp, c3[k] * kInvTemp };
      bool pid[4], pcam[4], isd[4];
#pragma unroll
      for (int s = 0; s < 4; ++s) {
        isd[s] = (rg[k] == cgv[s]);
        if (isd[s]) v[s] = kNegInf;
        pid[s]  = (rlab[k] == cl[s]);   // diag => pid true => excluded from negatives
        pcam[s] = pid[s] && (rcam[k] == cc[s]);
      }
      if (PASS == 0) {
        float tm = fmaxf(fmaxf(v[0], v[1]), fmaxf(v[2], v[3]));
        float nm = fmaxf(mx[k], tm);
        float sc = __expf(mx[k] - nm);   // one rescale per row per iteration
        float a1 = 0.f, a2 = 0.f;
#pragma unroll
        for (int s = 0; s < 4; ++s) {
          float ev = __expf(v[s] - nm);
          a1 += pid[s]  ? 0.f : ev;
          a2 += pcam[s] ? 0.f : ev;
        }
        s1[k] = s1[k] * sc + a1;
        s2[k] = s2[k] * sc + a2;
        mx[k] = nm;
      } else {
#pragma unroll
        for (int s = 0; s < 4; ++s) {
          bool pd = pid[s] && !isd[s];
          if (pd) {                      // positives are sparse: real branch
            float t = v[s] - mrow[k];
            float e = __expf(t);
            acc1[k] += __logf(e + n1r[k]) - t;
            if (pcam[s]) acc2[k] += __logf(e + n2r[k]) - t;
          }
        }
      }
    }
  }

  // Cross-lane (N-direction, 16 lanes per row) reduction.
  if (PASS == 0) {
#pragma unroll
    for (int k = 0; k < 8; ++k) {
      float m = mx[k], a = s1[k], b = s2[k];
#pragma unroll
      for (int off = 8; off >= 1; off >>= 1) {
        float om = __shfl_xor(m, off, 32);
        float oa = __shfl_xor(a, off, 32);
        float ob = __shfl_xor(b, off, 32);
        float nm = fmaxf(m, om);
        float sa = __expf(m - nm), sb = __expf(om - nm);
        a = a * sa + oa * sb;
        b = b * sa + ob * sb;
        m = nm;
      }
      if (l15 == 0) { mArr[rg[k]] = m; s1Arr[rg[k]] = a; s2Arr[rg[k]] = b; }
    }
  } else {
#pragma unroll
    for (int k = 0; k < 8; ++k) {
      float a = acc1[k], b = acc2[k];
#pragma unroll
      for (int off = 8; off >= 1; off >>= 1) {
        a += __shfl_xor(a, off, 32);
        b += __shfl_xor(b, off, 32);
      }
      if (l15 == 0)
        rowloss[rg[k]] = a / p1Arr[rg[k]] + 0.5f * b / p2Arr[rg[k]];
    }
  }
}

// ---------------- deterministic final reduction ----------------
__global__ __launch_bounds__(256) void final_reduce_kernel(const float* __restrict__ rowloss,
                                                           float* __restrict__ out) {
  __shared__ float sm[256];
  float a = 0.f;
  for (int r = threadIdx.x; r < kN2; r += 256) a += rowloss[r];
  sm[threadIdx.x] = a;
  __syncthreads();
  for (int s = 128; s > 0; s >>= 1) {
    if ((int)threadIdx.x < s) sm[threadIdx.x] += sm[threadIdx.x + s];
    __syncthreads();
  }
  if (threadIdx.x == 0) out[0] = sm[0] / (float)kN2;
}

extern "C" void kernel_launch(void* const* d_in, const int* in_sizes, int n_in,
                              void* d_out, int out_size, void* d_ws, size_t ws_size,
                              hipStream_t stream) {
  const float* feat  = (const float*)d_in[0];
  const int*   label = (const int*)d_in[1];
  const int*   camid = (const int*)d_in[2];
  float* out = (float*)d_out;

  // Workspace layout (floats): f[2N*D] | m | s1 | s2 | npos1 | npos2 | rowloss
  float* fw      = (float*)d_ws;
  float* mArr    = fw + (size_t)kN2 * kD;
  float* s1Arr   = mArr + kN2;
  float* s2Arr   = s1Arr + kN2;
  float* p1Arr   = s2Arr + kN2;
  float* p2Arr   = p1Arr + kN2;
  float* rowloss = p2Arr + kN2;

  pack_kernel<<<(kN2 * (kD / 4)) / 256, 256, 0, stream>>>(feat, fw);
  count_kernel<<<kN / 256, 256, 0, stream>>>(label, camid, p1Arr, p2Arr);
  scl_gemm_kernel<0><<<kN2 / kRowsWG, 256, 0, stream>>>(fw, label, camid, mArr, s1Arr,
                                                        s2Arr, p1Arr, p2Arr, rowloss);
  scl_gemm_kernel<1><<<kN2 / kRowsWG, 256, 0, stream>>>(fw, label, camid, mArr, s1Arr,
                                                        s2Arr, p1Arr, p2Arr, rowloss);
  final_reduce_kernel<<<1, 256, 0, stream>>>(rowloss, out);
}